// JobMarketGNN_38225208934803
// MI455X (gfx1250) — compile-verified
//
#include <hip/hip_runtime.h>

// GCN: 3 layers, N=50000 nodes, E=800000 edges, F_IN=HID=128, N_CLS=32.
// GEMMs: V_WMMA_F32_16X16X4_F32 with W staged in LDS, A tile held in VGPRs.
// Edge aggregation: wave-per-edge gather + fp32 atomics (L2-resident, the
// roofline bottleneck at ~1.5 GB total traffic).

typedef __attribute__((ext_vector_type(2))) float v2f;
typedef __attribute__((ext_vector_type(8))) float v8f;

// ---------------- degree / normalization ----------------

__global__ void deg_init_kernel(float* __restrict__ deg, int n) {
    int i = blockIdx.x * blockDim.x + threadIdx.x;
    if (i < n) deg[i] = 1.0f;          // self-loop contributes 1 to every node
}

__global__ void deg_count_kernel(const int* __restrict__ col, float* __restrict__ deg, int e) {
    int i = blockIdx.x * blockDim.x + threadIdx.x;
    if (i < e) atomicAdd(&deg[col[i]], 1.0f);
}

__global__ void deg_rsqrt_kernel(float* __restrict__ deg, int n) {
    int i = blockIdx.x * blockDim.x + threadIdx.x;
    if (i < n) deg[i] = rsqrtf(deg[i]);  // deg >= 1 always (self-loops)
}

// ---------------- dense GEMM via f32 WMMA, W staged in LDS ----------------
// C[M x NOUT] = op(A)[M x 128] * W[128 x NOUT], all row-major.
// Block = 256 threads = 8 waves. Block stages all of W into LDS (row-major:
// conflict-free since lanes index along n), then each wave computes one
// 16-row tile: full A tile in registers (32 x v2f), sweep NOUT/16 col tiles.
//
// Fragment layout per CDNA5 ISA 7.12.2 (V_WMMA_F32_16X16X4_F32):
//   A (16x4): lanes 0-15 -> row=lane, k={0,1}; lanes 16-31 -> row=lane-16, k={2,3}
//   B (4x16): lanes 0-15 -> n=lane, k={0,1};  lanes 16-31 -> n=lane-16, k={2,3}
//   C/D (16x16): VGPR r: lanes 0-15 -> row r, lanes 16-31 -> row r+8

template <int NOUT, bool RELU>
__global__ __launch_bounds__(256) void wmma_gemm_lds_kernel(
        const float* __restrict__ A,
        const float* __restrict__ W,
        float* __restrict__ C,
        int rowTiles) {
    constexpr int K = 128;
    extern __shared__ float wlds[];     // K * NOUT floats, row-major copy of W

    const int tid = threadIdx.x;
    for (int idx = tid; idx < K * NOUT; idx += 256)
        wlds[idx] = W[idx];
    __syncthreads();

    const int wave = tid >> 5;
    const int lane = tid & 31;
    const int half = lane >> 4;         // 0: lanes 0-15, 1: lanes 16-31
    const int l16  = lane & 15;

    const int mt = blockIdx.x * 8 + wave;
    if (mt >= rowTiles) return;         // wave-uniform exit (EXEC stays all-1s)
    const int m0 = mt * 16;

    // Load the wave's entire 16x128 A tile as fragments (64 VGPRs).
    v2f afrag[K / 4];
    const float* arow = A + (size_t)(m0 + l16) * K + half * 2;
#pragma unroll
    for (int s = 0; s < K / 4; ++s) {
        v2f a = *(const v2f*)(arow + s * 4);    // 8B-aligned pair
        if (RELU) { a.x = fmaxf(a.x, 0.f); a.y = fmaxf(a.y, 0.f); }
        afrag[s] = a;
    }

#pragma unroll
    for (int nt = 0; nt < NOUT / 16; ++nt) {
        const int n0 = nt * 16;
        v8f acc = {0.f, 0.f, 0.f, 0.f, 0.f, 0.f, 0.f, 0.f};
        const float* bbase = wlds + (size_t)(half * 2) * NOUT + (n0 + l16);
#pragma unroll
        for (int s = 0; s < K / 4; ++s) {
            v2f bv;
            bv.x = bbase[(size_t)(s * 4) * NOUT];         // W[k  ][n]
            bv.y = bbase[(size_t)(s * 4 + 1) * NOUT];     // W[k+1][n]
            // 8 args: (neg_a, A, neg_b, B, c_mod, C, reuse_a, reuse_b)
            acc = __builtin_amdgcn_wmma_f32_16x16x4_f32(false, afrag[s], false, bv,
                                                        (short)0, acc, false, false);
        }
        float* crow = C + (size_t)(m0 + half * 8) * NOUT + (n0 + l16);
#pragma unroll
        for (int r = 0; r < 8; ++r)
            crow[(size_t)r * NOUT] = acc[r];
    }
}

// ---------------- aggregation ----------------
// out[i][j] = h[i][j]*dinv[i]^2 + bias[j]   (self-loop term + bias)

template <int FDIM>
__global__ void agg_init_kernel(const float* __restrict__ h,
                                const float* __restrict__ dinv,
                                const float* __restrict__ bias,
                                float* __restrict__ out, int n) {
    int idx = blockIdx.x * blockDim.x + threadIdx.x;
    if (idx >= n * FDIM) return;
    int i = idx / FDIM;                 // FDIM is a power of two -> shifts
    int j = idx & (FDIM - 1);
    float d = dinv[i];
    out[idx] = h[idx] * d * d + bias[j];
}

// Edge scatter, F = 128: one wave per edge, each lane owns 4 consecutive floats.
__global__ void agg_edge128_kernel(const float* __restrict__ h,
                                   const int* __restrict__ row,
                                   const int* __restrict__ col,
                                   const float* __restrict__ dinv,
                                   float* __restrict__ out, int e) {
    int eid = blockIdx.x * (blockDim.x >> 5) + (threadIdx.x >> 5);
    if (eid >= e) return;
    int lane = threadIdx.x & 31;
    int r = row[eid];
    int c = col[eid];
    float w = dinv[r] * dinv[c];
    float4 v = ((const float4*)(h + (size_t)r * 128))[lane];
    float* op = out + (size_t)c * 128 + lane * 4;
    atomicAdd(op + 0, v.x * w);
    atomicAdd(op + 1, v.y * w);
    atomicAdd(op + 2, v.z * w);
    atomicAdd(op + 3, v.w * w);
}

// Edge scatter, F = 32: one wave per edge, one lane per feature.
__global__ void agg_edge32_kernel(const float* __restrict__ h,
                                  const int* __restrict__ row,
                                  const int* __restrict__ col,
                                  const float* __restrict__ dinv,
                                  float* __restrict__ out, int e) {
    int eid = blockIdx.x * (blockDim.x >> 5) + (threadIdx.x >> 5);
    if (eid >= e) return;
    int lane = threadIdx.x & 31;
    int r = row[eid];
    int c = col[eid];
    float w = dinv[r] * dinv[c];
    float v = h[(size_t)r * 32 + lane] * w;
    atomicAdd(out + (size_t)c * 32 + lane, v);
}

// ---------------- launcher ----------------

extern "C" void kernel_launch(void* const* d_in, const int* in_sizes, int n_in,
                              void* d_out, int out_size, void* d_ws, size_t ws_size,
                              hipStream_t stream) {
    const int F   = 128;
    const int CLS = 32;
    const int n = in_sizes[0] / F;      // 50000
    const int e = in_sizes[1] / 2;      // 800000

    const float* x   = (const float*)d_in[0];
    const int*   ei  = (const int*)d_in[1];
    const int*   row = ei;              // edge_index[0] = source
    const int*   col = ei + e;          // edge_index[1] = destination
    const float* W1  = (const float*)d_in[3];
    const float* b1  = (const float*)d_in[4];
    const float* W2  = (const float*)d_in[5];
    const float* b2  = (const float*)d_in[6];
    const float* W3  = (const float*)d_in[7];
    const float* b3  = (const float*)d_in[8];
    float*       out = (float*)d_out;

    // workspace: dinv[n] | bufh[n*128] | bufa[n*128]  (~51.4 MB)
    float* dinv = (float*)d_ws;
    float* bufh = dinv + n;
    float* bufa = bufh + (size_t)n * F;

    const int T = 256;
    const int wavesPerBlk = T / 32;
    dim3 blk(T);

    // --- normalization: deg -> dinv ---
    deg_init_kernel<<<(n + T - 1) / T, blk, 0, stream>>>(dinv, n);
    deg_count_kernel<<<(e + T - 1) / T, blk, 0, stream>>>(col, dinv, e);
    deg_rsqrt_kernel<<<(n + T - 1) / T, blk, 0, stream>>>(dinv, n);

    const int rowTiles  = n / 16;               // 3125
    const int gemmGrid  = (rowTiles + 7) / 8;   // 8 row-tiles (waves) per block
    const size_t ldsWide = (size_t)128 * F   * sizeof(float);   // 64 KB
    const size_t ldsCls  = (size_t)128 * CLS * sizeof(float);   // 16 KB

    // --- layer 1: h = x@W1 ; a = agg(h) + b1 (ReLU deferred into next GEMM) ---
    wmma_gemm_lds_kernel<128, false><<<gemmGrid, blk, ldsWide, stream>>>(x, W1, bufh, rowTiles);
    agg_init_kernel<128><<<((size_t)n * F + T - 1) / T, blk, 0, stream>>>(bufh, dinv, b1, bufa, n);
    agg_edge128_kernel<<<(e + wavesPerBlk - 1) / wavesPerBlk, blk, 0, stream>>>(bufh, row, col, dinv, bufa, e);

    // --- layer 2: h = relu(a)@W2 ; a = agg(h) + b2 ---
    wmma_gemm_lds_kernel<128, true><<<gemmGrid, blk, ldsWide, stream>>>(bufa, W2, bufh, rowTiles);
    agg_init_kernel<128><<<((size_t)n * F + T - 1) / T, blk, 0, stream>>>(bufh, dinv, b2, bufa, n);
    agg_edge128_kernel<<<(e + wavesPerBlk - 1) / wavesPerBlk, blk, 0, stream>>>(bufh, row, col, dinv, bufa, e);

    // --- layer 3: h = relu(a)@W3 ; out = agg(h) + b3 ---
    wmma_gemm_lds_kernel<32, true><<<gemmGrid, blk, ldsCls, stream>>>(bufa, W3, bufh, rowTiles);
    agg_init_kernel<32><<<((size_t)n * CLS + T - 1) / T, blk, 0, stream>>>(bufh, dinv, b3, out, n);
    agg_edge32_kernel<<<(e + wavesPerBlk - 1) / wavesPerBlk, blk, 0, stream>>>(bufh, row, col, dinv, out, e);
}